// VRNNGATA2C_74483322847928
// MI455X (gfx1250) — compile-verified
//
#include <hip/hip_runtime.h>

typedef float v2f __attribute__((ext_vector_type(2)));
typedef float v8f __attribute__((ext_vector_type(8)));

#define NEG_INF (-__builtin_inff())

static __device__ __forceinline__ float wave_sum(float s) {
#pragma unroll
  for (int off = 16; off; off >>= 1) s += __shfl_xor(s, off, 32);
  return s;
}
static __device__ __forceinline__ float wave_max(float s) {
#pragma unroll
  for (int off = 16; off; off >>= 1) s = fmaxf(s, __shfl_xor(s, off, 32));
  return s;
}

// ---------------------------------------------------------------------------
// Generic row-major f32 GEMM, one 16x16 C-tile per wave, K stepped by 4 via
// v_wmma_f32_16x16x4_f32. Layouts per CDNA5 ISA 7.12.2:
//   A 16x4 : lane L -> row M = L&15 ; half = L>>4 selects K pair {0,1}/{2,3}
//   B 4x16 : VGPR r, lanes 0-15 -> row K=r ; lanes 16-31 -> row K=r+2
//   C 16x16: VGPR r, lanes 0-15 -> row M=r ; lanes 16-31 -> row M=r+8
// blockDim = 128 (4 waves), grid = (N/64, M/16). N % 64 == 0, K % 4 == 0.
// ---------------------------------------------------------------------------
__global__ void __launch_bounds__(128)
vg_gemm_wmma(const float* __restrict__ A, const float* __restrict__ B,
             float* __restrict__ C, int M, int N, int K,
             int lda, int ldb, int ldc,
             const float* __restrict__ bias, int relu, int accum) {
  const int wave = threadIdx.x >> 5;
  const int lane = threadIdx.x & 31;
  const int tn = blockIdx.x * 4 + wave;   // tile col (wave-uniform)
  const int tm = blockIdx.y;              // tile row
  if (tn * 16 >= N) return;               // uniform across the wave
  const int l16 = lane & 15;
  const int half = lane >> 4;

  const float* __restrict__ Arow = A + (size_t)(tm * 16 + l16) * lda;
  const float* __restrict__ Bcol = B + tn * 16 + l16;

  v8f acc = {0.f, 0.f, 0.f, 0.f, 0.f, 0.f, 0.f, 0.f};
  for (int k = 0; k < K; k += 4) {
    v2f a;
    a.x = Arow[k + half * 2 + 0];
    a.y = Arow[k + half * 2 + 1];
    v2f b;
    b.x = Bcol[(size_t)(k + half * 2 + 0) * ldb];
    b.y = Bcol[(size_t)(k + half * 2 + 1) * ldb];
    acc = __builtin_amdgcn_wmma_f32_16x16x4_f32(
        false, a, false, b, (short)0, acc, false, false);
  }

  const int col = tn * 16 + l16;
#pragma unroll
  for (int r = 0; r < 8; ++r) {
    const int row = tm * 16 + half * 8 + r;
    float v = acc[r];
    if (accum) v += C[(size_t)row * ldc + col];
    if (bias)  v += bias[col];
    if (relu)  v = fmaxf(v, 0.f);
    C[(size_t)row * ldc + col] = v;
  }
}

// out[i] = M[i,:] . (va + vb)   (vb optional). One wave per row.
__global__ void __launch_bounds__(256)
vg_matvec(const float* __restrict__ Mm, const float* __restrict__ va,
          const float* __restrict__ vb, float* __restrict__ out,
          int rows, int cols) {
  const int w = (blockIdx.x * blockDim.x + threadIdx.x) >> 5;
  const int lane = threadIdx.x & 31;
  if (w >= rows) return;
  const float* row = Mm + (size_t)w * cols;
  float s = 0.f;
  for (int k = lane; k < cols; k += 32) {
    float wt = va[k] + (vb ? vb[k] : 0.f);
    s += row[k] * wt;
  }
  s = wave_sum(s);
  if (lane == 0) out[w] = s;
}

// P[i,j] = softmax_j( adj? u_i+v_j : -inf ).  One block (256 thr) per row.
__global__ void __launch_bounds__(256)
vg_attn_softmax(const float* __restrict__ u, const float* __restrict__ v,
                const int* __restrict__ adj, float* __restrict__ P, int n) {
  const int i = blockIdx.x, t = threadIdx.x;
  __shared__ float red[256];
  const float ui = u[i];
  float s0 = (adj[(size_t)i * n + t]       != 0) ? (ui + v[t])       : NEG_INF;
  float s1 = (adj[(size_t)i * n + t + 256] != 0) ? (ui + v[t + 256]) : NEG_INF;
  red[t] = fmaxf(s0, s1); __syncthreads();
  for (int st = 128; st; st >>= 1) {
    if (t < st) red[t] = fmaxf(red[t], red[t + st]);
    __syncthreads();
  }
  const float m = red[0]; __syncthreads();
  const float e0 = expf(s0 - m), e1 = expf(s1 - m);
  red[t] = e0 + e1; __syncthreads();
  for (int st = 128; st; st >>= 1) {
    if (t < st) red[t] += red[t + st];
    __syncthreads();
  }
  const float inv = 1.f / red[0];
  P[(size_t)i * n + t]       = e0 * inv;
  P[(size_t)i * n + t + 256] = e1 * inv;
}

// w2a[k] = ce_w2[k,:].(a0+a1) ; b2a = ce_b2.(a0+a1)
__global__ void __launch_bounds__(256)
vg_w2a(const float* __restrict__ ce_w2, const float* __restrict__ ce_b2,
       const float* __restrict__ a0, const float* __restrict__ a1,
       float* __restrict__ w2a, float* __restrict__ b2a) {
  const int k = threadIdx.x;
  float s = 0.f;
  for (int m = 0; m < 64; ++m) s += ce_w2[k * 64 + m] * (a0[m] + a1[m]);
  w2a[k] = s;
  if (k == 0) {
    float b = 0.f;
    for (int m = 0; m < 64; ++m) b += ce_b2[m] * (a0[m] + a1[m]);
    b2a[0] = b;
  }
}

// Pass 1: cs[i,j] = relu(A_i+B_j+b1).w2a + b2a  (diag -> -inf); per-block max.
__global__ void __launch_bounds__(256)
vg_pair_pass1(const float* __restrict__ Am, const float* __restrict__ Bm,
              const float* __restrict__ b1, const float* __restrict__ w2a,
              const float* __restrict__ b2a, float* __restrict__ cs,
              float* __restrict__ blockmax, int n) {
  const int wave = threadIdx.x >> 5, lane = threadIdx.x & 31;
  const int i = blockIdx.x * 8 + wave;
  float a[8], w[8];
#pragma unroll
  for (int t = 0; t < 8; ++t) {
    const int k = lane * 8 + t;
    a[t] = Am[(size_t)i * 256 + k] + b1[k];
    w[t] = w2a[k];
  }
  const float bb = b2a[0];
  float wmax = NEG_INF;
  for (int j = 0; j < n; ++j) {
    if (j + 1 < n) __builtin_prefetch(Bm + (size_t)(j + 1) * 256 + lane * 8, 0, 0);
    float s = 0.f;
#pragma unroll
    for (int t = 0; t < 8; ++t) {
      const float x = a[t] + Bm[(size_t)j * 256 + lane * 8 + t];
      s += fmaxf(x, 0.f) * w[t];
    }
    s = wave_sum(s) + bb;
    if (i == j) s = NEG_INF;
    if (lane == 0) cs[(size_t)i * n + j] = s;
    wmax = fmaxf(wmax, s);
  }
  __shared__ float sm[8];
  if (lane == 0) sm[wave] = wmax;
  __syncthreads();
  if (threadIdx.x == 0) {
    float m = sm[0];
    for (int t = 1; t < 8; ++t) m = fmaxf(m, sm[t]);
    blockmax[blockIdx.x] = m;
  }
}

__global__ void __launch_bounds__(64)
vg_reduce_max64(const float* __restrict__ bm, float* __restrict__ Mx) {
  __shared__ float sm[64];
  const int t = threadIdx.x;
  sm[t] = bm[t]; __syncthreads();
  for (int st = 32; st; st >>= 1) {
    if (t < st) sm[t] = fmaxf(sm[t], sm[t + st]);
    __syncthreads();
  }
  if (t == 0) Mx[0] = sm[0];
}

// Pass 2: p = exp(cs - max); S += p*relu(A_i+B_j+b1); denom += p.
__global__ void __launch_bounds__(256)
vg_pair_pass2(const float* __restrict__ Am, const float* __restrict__ Bm,
              const float* __restrict__ b1, const float* __restrict__ cs,
              const float* __restrict__ Mx, float* __restrict__ S,
              float* __restrict__ denom, int n) {
  const int wave = threadIdx.x >> 5, lane = threadIdx.x & 31;
  const int i = blockIdx.x * 8 + wave;
  float a[8], Sl[8];
#pragma unroll
  for (int t = 0; t < 8; ++t) {
    a[t] = Am[(size_t)i * 256 + lane * 8 + t] + b1[lane * 8 + t];
    Sl[t] = 0.f;
  }
  const float Mv = Mx[0];
  float dl = 0.f;
  for (int j = 0; j < n; ++j) {
    const float c = cs[(size_t)i * n + j];
    const float e = (i == j) ? 0.f : expf(c - Mv);
    dl += e;
#pragma unroll
    for (int t = 0; t < 8; ++t) {
      const float x = a[t] + Bm[(size_t)j * 256 + lane * 8 + t];
      Sl[t] += e * fmaxf(x, 0.f);
    }
  }
#pragma unroll
  for (int t = 0; t < 8; ++t) atomicAdd(&S[lane * 8 + t], Sl[t]);
  if (lane == 0) atomicAdd(denom, dl);
}

// h2vec[m] = (S/denom).ce_w2[:,m] + ce_b2[m]
__global__ void __launch_bounds__(64)
vg_h2(const float* __restrict__ S, const float* __restrict__ denom,
      const float* __restrict__ ce_w2, const float* __restrict__ ce_b2,
      float* __restrict__ h2vec) {
  const int m = threadIdx.x;
  const float inv = 1.f / denom[0];
  float s = 0.f;
  for (int k = 0; k < 256; ++k) s += S[k] * ce_w2[k * 64 + m];
  h2vec[m] = s * inv + ce_b2[m];
}

// Xt = [V | prev_hidden]  (512 x 512)
__global__ void __launch_bounds__(256)
vg_concat2(const float* __restrict__ V, const float* __restrict__ Ph,
           float* __restrict__ Xt) {
  const int idx = blockIdx.x * 256 + threadIdx.x;
  const int row = idx >> 9, col = idx & 511;
  Xt[idx] = (col < 256) ? V[row * 256 + col] : Ph[row * 256 + col - 256];
}

// H3[i,:] = mean(tf[0..i,:])   (causal softmax over a broadcast column is uniform)
__global__ void __launch_bounds__(256)
vg_h3_mean(const float* __restrict__ tf, float* __restrict__ H3) {
  const int wave = threadIdx.x >> 5, lane = threadIdx.x & 31;
  const int i = blockIdx.x * 8 + wave;
  float a0 = 0.f, a1 = 0.f;
  for (int j = 0; j <= i; ++j) {
    a0 += tf[j * 64 + lane];
    a1 += tf[j * 64 + lane + 32];
  }
  const float inv = 1.f / (float)(i + 1);
  H3[i * 64 + lane]      = a0 * inv;
  H3[i * 64 + lane + 32] = a1 * inv;
}

// First two columns j with adj[i,j]==1 (stable argsort of 1-adj), degree>=2 flag.
__global__ void __launch_bounds__(256)
vg_neighbors(const int* __restrict__ adj, int* __restrict__ nbr, int n) {
  const int i = blockIdx.x * 256 + threadIdx.x;
  if (i >= n) return;
  int j0 = 0, j1 = 0, cnt = 0;
  for (int j = 0; j < n; ++j) {
    if (adj[(size_t)i * n + j] != 0) {
      if (cnt == 0) j0 = j; else if (cnt == 1) j1 = j;
      ++cnt;
    }
  }
  nbr[i * 3 + 0] = j0;
  nbr[i * 3 + 1] = j1;
  nbr[i * 3 + 2] = (cnt >= 2) ? 1 : 0;
}

// Ci = [V_i | V_j0 or 0 | V_j1 or 0]   (512 x 768)
__global__ void __launch_bounds__(256)
vg_gather_ci(const float* __restrict__ V, const int* __restrict__ nbr,
             float* __restrict__ Ci) {
  const int idx = blockIdx.x * 256 + threadIdx.x;
  const int i = idx / 768, c = idx % 768;
  float val;
  if (c < 256) {
    val = V[i * 256 + c];
  } else {
    const int which = (c - 256) >> 8;
    const int cc = (c - 256) & 255;
    val = nbr[i * 3 + 2] ? V[nbr[i * 3 + which] * 256 + cc] : 0.f;
  }
  Ci[idx] = val;
}

// h4vec[m] = sum_i cf4[i,m]
__global__ void __launch_bounds__(64)
vg_colsum(const float* __restrict__ cf4, float* __restrict__ h4vec, int rows) {
  const int m = threadIdx.x;
  float s = 0.f;
  for (int i = 0; i < rows; ++i) s += cf4[i * 64 + m];
  h4vec[m] = s;
}

// c2[n] = h2vec . W2[256:320, n]
__global__ void __launch_bounds__(256)
vg_c2(const float* __restrict__ h2vec, const float* __restrict__ W2,
      float* __restrict__ c2) {
  const int nn = threadIdx.x;
  float s = 0.f;
  for (int m = 0; m < 64; ++m) s += h2vec[m] * W2[(256 + m) * 256 + nn];
  c2[nn] = s;
}

// Hc[i,n] += c2[n] + (i<64 ? h4vec[i]*W2[384,n] : 0)
__global__ void __launch_bounds__(256)
vg_hc_finish(float* __restrict__ Hc, const float* __restrict__ c2,
             const float* __restrict__ h4vec, const float* __restrict__ w2r384) {
  const int idx = blockIdx.x * 256 + threadIdx.x;
  const int i = idx >> 8, nn = idx & 255;
  float v = Hc[idx] + c2[nn];
  if (i < 64) v += h4vec[i] * w2r384[nn];
  Hc[idx] = v;
}

// Row LayerNorm(256) + ELU -> out
__global__ void __launch_bounds__(256)
vg_ln_elu(const float* __restrict__ X, const float* __restrict__ g,
          const float* __restrict__ b, float* __restrict__ out) {
  const int i = blockIdx.x, t = threadIdx.x;
  __shared__ float red[256];
  const float x = X[i * 256 + t];
  red[t] = x; __syncthreads();
  for (int st = 128; st; st >>= 1) { if (t < st) red[t] += red[t + st]; __syncthreads(); }
  const float mu = red[0] * (1.f / 256.f); __syncthreads();
  const float d = x - mu;
  red[t] = d * d; __syncthreads();
  for (int st = 128; st; st >>= 1) { if (t < st) red[t] += red[t + st]; __syncthreads(); }
  const float var = red[0] * (1.f / 256.f);
  const float y = d * rsqrtf(var + 1e-5f) * g[t] + b[t];
  out[i * 256 + t] = (y > 0.f) ? y : expm1f(y);
}

// ---------------------------------------------------------------------------
extern "C" void kernel_launch(void* const* d_in, const int* in_sizes, int n_in,
                              void* d_out, int out_size, void* d_ws, size_t ws_size,
                              hipStream_t stream) {
  (void)in_sizes; (void)n_in; (void)out_size; (void)ws_size;
  const int N = 512;

  const float* V    = (const float*)d_in[0];
  const int*   adj  = (const int*)  d_in[1];
  const float* Ph   = (const float*)d_in[2];
  const float* W1   = (const float*)d_in[3];
  const float* a_s0 = (const float*)d_in[4];
  const float* a_s1 = (const float*)d_in[5];
  const float* cew1 = (const float*)d_in[6];
  const float* ceb1 = (const float*)d_in[7];
  const float* cew2 = (const float*)d_in[8];
  const float* ceb2 = (const float*)d_in[9];
  const float* a_c0 = (const float*)d_in[10];
  const float* a_c1 = (const float*)d_in[11];
  const float* tew1 = (const float*)d_in[12];
  const float* teb1 = (const float*)d_in[13];
  const float* tew2 = (const float*)d_in[14];
  const float* teb2 = (const float*)d_in[15];
  const float* cow1 = (const float*)d_in[18];
  const float* cob1 = (const float*)d_in[19];
  const float* cow2 = (const float*)d_in[20];
  const float* cob2 = (const float*)d_in[21];
  const float* W2   = (const float*)d_in[24];
  const float* Wo   = (const float*)d_in[25];
  const float* bo   = (const float*)d_in[26];
  const float* lng  = (const float*)d_in[27];
  const float* lnb  = (const float*)d_in[28];
  float* out = (float*)d_out;

  float* w = (float*)d_ws;
  size_t o = 0;
  auto alloc = [&](size_t nf) { float* p = w + o; o += nf; return p; };
  float* Wh1  = alloc(512 * 256);
  float* P    = alloc(512 * 512);
  float* H1   = alloc(512 * 256);
  float* Am   = alloc(512 * 256);
  float* Bm   = alloc(512 * 256);
  float* cs   = alloc(512 * 512);
  float* bmax = alloc(64);
  float* Mx   = alloc(64);       // scalar, padded
  float* S    = alloc(256);
  float* den  = alloc(64);       // scalar, padded
  float* h2v  = alloc(64);
  float* w2a  = alloc(256);
  float* b2a  = alloc(64);
  float* uvec = alloc(512);
  float* vvec = alloc(512);
  float* Xt   = alloc(512 * 512);
  float* T1   = alloc(512 * 256);
  float* tf   = alloc(512 * 64);
  float* H3   = alloc(512 * 64);
  float* Ci   = alloc(512 * 768);
  float* C1   = alloc(512 * 256);
  float* cf4  = alloc(512 * 64);
  float* h4v  = alloc(64);
  float* c2   = alloc(256);
  float* Hc   = alloc(512 * 256);
  float* Ob   = alloc(512 * 256);
  int*   nbr  = (int*)alloc(512 * 3 + 64);

  const dim3 g256(256 / 64, 512 / 16), g64(64 / 64, 512 / 16);

  // --- branch 1: GAT attention ---
  vg_gemm_wmma<<<g256, 128, 0, stream>>>(V, W1, Wh1, 512, 256, 256, 256, 256, 256, nullptr, 0, 0);
  vg_matvec<<<64, 256, 0, stream>>>(Wh1, a_s0, nullptr, uvec, 512, 256);
  vg_matvec<<<64, 256, 0, stream>>>(Wh1, a_s1, nullptr, vvec, 512, 256);
  vg_attn_softmax<<<512, 256, 0, stream>>>(uvec, vvec, adj, P, N);
  vg_gemm_wmma<<<g256, 128, 0, stream>>>(P, Wh1, H1, 512, 256, 512, 512, 256, 256, nullptr, 0, 0);

  // --- branch 2: pairwise edge MLP, folded ---
  vg_gemm_wmma<<<g256, 128, 0, stream>>>(V, cew1,             Am, 512, 256, 256, 256, 256, 256, nullptr, 0, 0);
  vg_gemm_wmma<<<g256, 128, 0, stream>>>(V, cew1 + 256 * 256, Bm, 512, 256, 256, 256, 256, 256, nullptr, 0, 0);
  vg_w2a<<<1, 256, 0, stream>>>(cew2, ceb2, a_c0, a_c1, w2a, b2a);
  vg_pair_pass1<<<64, 256, 0, stream>>>(Am, Bm, ceb1, w2a, b2a, cs, bmax, N);
  vg_reduce_max64<<<1, 64, 0, stream>>>(bmax, Mx);
  hipMemsetAsync(S, 0, 256 * sizeof(float), stream);
  hipMemsetAsync(den, 0, sizeof(float), stream);
  vg_pair_pass2<<<64, 256, 0, stream>>>(Am, Bm, ceb1, cs, Mx, S, den, N);
  vg_h2<<<1, 64, 0, stream>>>(S, den, cew2, ceb2, h2v);

  // --- branch 3: temporal MLP + (degenerate) causal softmax ---
  vg_concat2<<<(512 * 512) / 256, 256, 0, stream>>>(V, Ph, Xt);
  vg_gemm_wmma<<<g256, 128, 0, stream>>>(Xt, tew1, T1, 512, 256, 512, 512, 256, 256, teb1, 1, 0);
  vg_gemm_wmma<<<g64, 128, 0, stream>>>(T1, tew2, tf, 512, 64, 256, 256, 64, 64, teb2, 0, 0);
  vg_h3_mean<<<64, 256, 0, stream>>>(tf, H3);

  // --- branch 4: neighbor MLP, column-summed ---
  vg_neighbors<<<2, 256, 0, stream>>>(adj, nbr, N);
  vg_gather_ci<<<(512 * 768) / 256, 256, 0, stream>>>(V, nbr, Ci);
  vg_gemm_wmma<<<g256, 128, 0, stream>>>(Ci, cow1, C1, 512, 256, 768, 768, 256, 256, cob1, 1, 0);
  vg_gemm_wmma<<<g64, 128, 0, stream>>>(C1, cow2, cf4, 512, 64, 256, 256, 64, 64, cob2, 0, 0);
  vg_colsum<<<1, 64, 0, stream>>>(cf4, h4v, 512);

  // --- combine: Hc = H1@W2[:256] + H3@W2[320:384] + c2 + h4 rank-1 ---
  vg_c2<<<1, 256, 0, stream>>>(h2v, W2, c2);
  vg_gemm_wmma<<<g256, 128, 0, stream>>>(H1, W2, Hc, 512, 256, 256, 256, 256, 256, nullptr, 0, 0);
  vg_gemm_wmma<<<g256, 128, 0, stream>>>(H3, W2 + 320 * 256, Hc, 512, 256, 64, 64, 256, 256, nullptr, 0, 1);
  vg_hc_finish<<<(512 * 256) / 256, 256, 0, stream>>>(Hc, c2, h4v, W2 + 384 * 256);

  // --- output projection + LayerNorm + ELU ---
  vg_gemm_wmma<<<g256, 128, 0, stream>>>(Hc, Wo, Ob, 512, 256, 256, 256, 256, 256, bo, 0, 0);
  vg_ln_elu<<<512, 256, 0, stream>>>(Ob, lng, lnb, out);
}